// GlobalSphericalBasisLayer_48627619726062
// MI455X (gfx1250) — compile-verified
//
#include <hip/hip_runtime.h>

// GlobalSphericalBasisLayer fused kernel for MI455X (gfx1250, wave32).
// Memory-bound (~625 MB traffic vs ~10 GFLOP): one fused pass, fp32 WMMA
// (v_wmma_f32_16x16x4_f32) for the sph(16x36) @ kernel(36x64) projection,
// non-temporal stores for the 512 MB output and the int64 index passthrough.
//
// Round-1 fix: avoid select-of-pointer on the sph[] array (it defeated SROA
// and PromoteAlloca pushed the array into LDS -> ds_load/ds_store churn).
// Now: constant-index scalar loads, then value selects; A fragments built
// once and reused across all 4 N-tiles.

#define LMAX 6
#define NSPH 36   // LMAX*LMAX
#define EDIM 64

typedef __attribute__((ext_vector_type(2))) float     v2f;
typedef __attribute__((ext_vector_type(8))) float     v8f;
typedef __attribute__((ext_vector_type(2))) long long v2i64;

// ---------- compile-time spherical-harmonic normalization constants ----------
constexpr double kPi = 3.14159265358979323846;
constexpr double cfact(int n)  { double r = 1.0; for (int i = 2; i <= n; ++i)  r *= i; return r; }
constexpr double cdfact(int n) { double r = 1.0; for (int i = 1; i <= n; i += 2) r *= i; return r; }
// constexpr sqrt via Newton iteration (portable: no reliance on constexpr __builtin_sqrt)
constexpr double csqrt(double x) {
  double r = x > 1.0 ? x : 1.0;
  for (int i = 0; i < 100; ++i) r = 0.5 * (r + x / r);
  return r;
}

struct CoefTab {
  float c[NSPH];
  constexpr CoefTab() : c{} {
    int idx = 0;
    for (int l = 0; l < LMAX; ++l)
      for (int m = -l; m <= l; ++m) {
        int am = m < 0 ? -m : m;
        double K = csqrt((2.0 * l + 1.0) / (4.0 * kPi) * cfact(l - am) / cfact(l + am));
        c[idx++] = (float)((m == 0) ? K : csqrt(2.0) * K);
      }
  }
};
constexpr CoefTab kCoef{};

struct DfTab {
  float d[LMAX];
  constexpr DfTab() : d{} { for (int m = 0; m < LMAX; ++m) d[m] = (float)cdfact(2 * m - 1); }
};
constexpr DfTab kDf{};

// ---------- real spherical harmonics, fully unrolled into registers ----------
__device__ __forceinline__ void eval_sph(float x, float s, float cp, float sp,
                                         float sph[NSPH]) {
  // x = cos(theta), s = sin(theta) (>=0); cp = cos(phi), sp = sin(phi) (>=0)
  float P[LMAX][LMAX];
  P[0][0] = 1.f;
  {
    float sm = 1.f;
#pragma unroll
    for (int m = 1; m < LMAX; ++m) {
      sm *= s;
      P[m][m] = ((m & 1) ? -kDf.d[m] : kDf.d[m]) * sm;   // (-1)^m (2m-1)!! s^m
    }
  }
#pragma unroll
  for (int m = 0; m < LMAX - 1; ++m)
    P[m + 1][m] = (2.f * m + 1.f) * x * P[m][m];
#pragma unroll
  for (int m = 0; m < LMAX; ++m)
#pragma unroll
    for (int l = m + 2; l < LMAX; ++l)
      P[l][m] = ((2.f * l - 1.f) * x * P[l - 1][m] - (l + m - 1.f) * P[l - 2][m]) *
                (1.f / (float)(l - m));

  // cos(m*phi), sin(m*phi) by Chebyshev-style recurrence
  float cmv[LMAX], smv[LMAX];
  cmv[0] = 1.f; smv[0] = 0.f;
#pragma unroll
  for (int m = 1; m < LMAX; ++m) {
    cmv[m] = cmv[m - 1] * cp - smv[m - 1] * sp;
    smv[m] = smv[m - 1] * cp + cmv[m - 1] * sp;
  }

  int idx = 0;
#pragma unroll
  for (int l = 0; l < LMAX; ++l) {
#pragma unroll
    for (int m = -l; m <= l; ++m) {
      int am = m < 0 ? -m : m;
      float v = kCoef.c[idx] * P[l][am];
      if (m < 0)      v *= smv[am];
      else if (m > 0) v *= cmv[am];
      sph[idx] = v;
      ++idx;
    }
  }
}

// ---------- fused sph + projection kernel ----------
// Block = 256 threads = 8 waves; each wave owns one 16-edge tile.
// Lanes 16..31 duplicate lanes 0..15's edge so the fp32 WMMA A-operand layout
// (K=0,1 in lanes 0-15 / K=2,3 in lanes 16-31, same M rows) falls out of
// registers with no cross-lane traffic.
__global__ __launch_bounds__(256) void sph_edge_wmma(
    const float* __restrict__ env, const float* __restrict__ kern,
    const float* __restrict__ bias, float* __restrict__ out, long long E) {
  const int lane = threadIdx.x & 31;
  const int wave = threadIdx.x >> 5;
  const long long ntiles = (E + 15) >> 4;
  const long long tile = (long long)blockIdx.x * 8 + wave;
  if (tile >= ntiles) return;  // whole wave exits together -> EXEC stays all-1s

  const bool hi = (lane >= 16);
  long long e = tile * 16 + (lane & 15);
  long long eld = e < E ? e : E - 1;

  const float* pe = env + eld * 6;
  float p0 = pe[0], p1 = pe[1], p2 = pe[2];
  float e0 = pe[3], e1 = pe[4], e2 = pe[5];

  // theta = atan2(||p x e||, p.e)  -> need cos/sin only
  float dot_t = p0 * e0 + p1 * e1 + p2 * e2;
  float cx = p1 * e2 - p2 * e1;
  float cy = p2 * e0 - p0 * e2;
  float cz = p0 * e1 - p1 * e0;
  float cross_t = sqrtf(cx * cx + cy * cy + cz * cz);
  float rt = sqrtf(dot_t * dot_t + cross_t * cross_t);
  float ct = rt > 0.f ? dot_t / rt : 1.f;    // atan2(0,0)=0 => cos=1
  float st = rt > 0.f ? cross_t / rt : 0.f;

  // edges_proj = e * cos(theta); phi = atan2(||p x e_proj||, e_proj.z)
  float dot_p = e2 * ct;
  float cross_p = fabsf(ct) * cross_t;       // ||ct * (p x e)||
  float rp = sqrtf(dot_p * dot_p + cross_p * cross_p);
  float cp = rp > 0.f ? dot_p / rp : 1.f;
  float sp = rp > 0.f ? cross_p / rp : 0.f;

  float sph[NSPH];
  eval_sph(ct, st, cp, sp, sph);

  // Build the 9 A-operand fragments ONCE (shared by all 4 N-tiles).
  // All sph[] indices are compile-time constants; the hi/lo choice is a
  // value select (v_cndmask on registers), never a pointer select, so the
  // sph alloca stays fully SROA'd in VGPRs.
  v2f afrag[9];
#pragma unroll
  for (int kk = 0; kk < 9; ++kk) {
    const float a0 = sph[kk * 4 + 0];
    const float a1 = sph[kk * 4 + 1];
    const float a2 = sph[kk * 4 + 2];
    const float a3 = sph[kk * 4 + 3];
    v2f a;
    a.x = hi ? a2 : a0;
    a.y = hi ? a3 : a1;
    afrag[kk] = a;
  }

  const int n0 = lane & 15;
  const int koff = hi ? 2 : 0;

#pragma unroll
  for (int nt = 0; nt < 4; ++nt) {
    const int n = n0 + nt * 16;
    const float bv = bias[n];
    v8f acc = {bv, bv, bv, bv, bv, bv, bv, bv};
#pragma unroll
    for (int kk = 0; kk < 9; ++kk) {          // K = 36 = 9 * 4
      v2f b;
      b.x = kern[(kk * 4 + koff + 0) * EDIM + n];     // 9 KB matrix: lives in cache
      b.y = kern[(kk * 4 + koff + 1) * EDIM + n];
      acc = __builtin_amdgcn_wmma_f32_16x16x4_f32(
          /*neg_a=*/false, afrag[kk], /*neg_b=*/false, b,
          /*c_mod=*/(short)0, acc, /*reuse_a=*/false, /*reuse_b=*/false);
    }
    // C/D layout: VGPR r -> row r (lanes 0-15) / row r+8 (lanes 16-31), col = lane&15
#pragma unroll
    for (int r = 0; r < 8; ++r) {
      long long row = tile * 16 + r + (hi ? 8 : 0);
      if (row < E)
        __builtin_nontemporal_store(acc[r], &out[row * EDIM + n]);  // 512MB, never re-read
    }
  }
}

// ---------- int64 pair_indices passthrough (16 B per edge) ----------
__global__ void copy_pair_idx(const v2i64* __restrict__ src, v2i64* __restrict__ dst,
                              long long n) {
  long long i = (long long)blockIdx.x * blockDim.x + threadIdx.x;
  if (i < n) {
    v2i64 v = __builtin_nontemporal_load(&src[i]);
    __builtin_nontemporal_store(v, &dst[i]);
  }
}

extern "C" void kernel_launch(void* const* d_in, const int* in_sizes, int n_in,
                              void* d_out, int out_size, void* d_ws, size_t ws_size,
                              hipStream_t stream) {
  const float*     env  = (const float*)d_in[0];
  const long long* pidx = (const long long*)d_in[1];
  const float*     kern = (const float*)d_in[2];
  const float*     bias = (const float*)d_in[3];
  float*           out  = (float*)d_out;

  const long long E = (long long)in_sizes[0] / 6;

  const long long tiles = (E + 15) / 16;
  const int blocks = (int)((tiles + 7) / 8);
  sph_edge_wmma<<<blocks, 256, 0, stream>>>(env, kern, bias, out, E);

  // pair_indices appended flat after the (E,64) fp32 output; E*64*4 is 16B-aligned
  v2i64* dst = (v2i64*)(out + E * (long long)EDIM);
  const int cblocks = (int)((E + 255) / 256);
  copy_pair_idx<<<cblocks, 256, 0, stream>>>((const v2i64*)pidx, dst, E);
}